// DistanceLoss_90606630076684
// MI455X (gfx1250) — compile-verified
//
#include <hip/hip_runtime.h>

typedef float v2f __attribute__((ext_vector_type(2)));
typedef float v8f __attribute__((ext_vector_type(8)));

// ---------------------------------------------------------------------------
// Pre-pack kernels: build WMMA-ready operand layouts in workspace.
//   queries -> (-2x, -2y, -2z, x^2+y^2+z^2)
//   points  -> ( x,   y,   z,  x^2+y^2+z^2)
// ---------------------------------------------------------------------------
__global__ void nn_pack_queries(const float* __restrict__ q,
                                float4* __restrict__ o, int total) {
    int i = blockIdx.x * blockDim.x + threadIdx.x;
    if (i < total) {
        float x = q[3 * i + 0], y = q[3 * i + 1], z = q[3 * i + 2];
        o[i] = make_float4(-2.0f * x, -2.0f * y, -2.0f * z,
                           x * x + y * y + z * z);
    }
}

__global__ void nn_pack_points(const float* __restrict__ p,
                               float4* __restrict__ o, int total) {
    int i = blockIdx.x * blockDim.x + threadIdx.x;
    if (i < total) {
        float x = p[3 * i + 0], y = p[3 * i + 1], z = p[3 * i + 2];
        o[i] = make_float4(x, y, z, x * x + y * y + z * z);
    }
}

// ---------------------------------------------------------------------------
// Main kernel: one wave32 owns 16 queries of one batch, sweeps all M points
// in 16-point tiles. Each tile is ONE v_wmma_f32_16x16x4_f32 producing the
// full 16x16 squared-distance tile (identity folded into A/B/C operands:
// A row = [-2x,-2y,-2z,1], B col = [x,y,z,||p||^2], C row = ||q||^2),
// followed by 8 v_min into the running accumulator (compiler fuses to min3).
// ---------------------------------------------------------------------------
__global__ __launch_bounds__(256)
void nn_wmma_kernel(const float4* __restrict__ qpack,
                    const float4* __restrict__ ppack,
                    float* __restrict__ out,
                    int N, int M, int blocksPerBatch) {
    const int b    = blockIdx.x / blocksPerBatch;
    const int blk  = blockIdx.x % blocksPerBatch;
    const int wave = threadIdx.x >> 5;
    const int lane = threadIdx.x & 31;
    const int half = lane >> 4;    // 0: K=0,1 lanes   1: K=2,3 lanes
    const int lm   = lane & 15;
    const int qbase = blk * 128 + wave * 16;   // 16 queries per wave

    const float4* qrow = qpack + (size_t)b * N + qbase;
    const float4* prow = ppack + (size_t)b * M;

    // A fragment (16x4): row = lm. lanes 0-15 hold K=0,1; lanes 16-31 K=2,3.
    float4 q = qrow[lm];
    v2f A;
    A.x = half ? q.z : q.x;     // K=0 -> -2x ; K=2 -> -2z
    A.y = half ? 1.0f : q.y;    // K=1 -> -2y ; K=3 -> 1

    // C fragment (16x16): C[i][*] = ||q_i||^2. VGPR r holds row r (lanes 0-15)
    // and row 8+r (lanes 16-31).
    v8f C;
#pragma unroll
    for (int r = 0; r < 8; ++r) C[r] = qrow[half * 8 + r].w;

    v8f acc;
#pragma unroll
    for (int r = 0; r < 8; ++r) acc[r] = 3.0e38f;

    // B fragment source: lane pair (lm, lm+16) reads the two halves of
    // pts4[tile*16 + lm] -> wave reads 256 contiguous bytes per tile.
    const char* pb = (const char*)prow + (lm * 16 + half * 8);
    const int tiles = M >> 4;

#pragma unroll 4
    for (int t = 0; t < tiles; ++t) {
        v2f Bf = *(const v2f*)(pb + (size_t)t * 256);
        // High-locality hint -> WGP-scope prefetch (pull line into WGP$/L0).
        __builtin_prefetch(pb + (size_t)(t + 16) * 256, 0, 3);
        // D = A x B + C  ==  full 16x16 tile of squared distances
        v8f d = __builtin_amdgcn_wmma_f32_16x16x4_f32(
            /*neg_a=*/false, A, /*neg_b=*/false, Bf,
            /*c_mod=*/(short)0, C, /*reuse_a=*/false, /*reuse_b=*/false);
#pragma unroll
        for (int r = 0; r < 8; ++r) acc[r] = fminf(acc[r], d[r]);
    }

    // Min-reduce across the 16 columns (lanes within each 16-lane half).
    // XOR masks < 16 never cross the half boundary in wave32.
#pragma unroll
    for (int off = 1; off < 16; off <<= 1) {
#pragma unroll
        for (int r = 0; r < 8; ++r) {
            float o = __shfl_xor(acc[r], off, 32);
            acc[r] = fminf(acc[r], o);
        }
    }

    // Lane 0 holds rows 0..7, lane 16 rows 8..15. Clamp (monotone, commutes
    // with min) and store.
    if (lm == 0) {
        float* orow = out + (size_t)b * N + qbase + half * 8;
#pragma unroll
        for (int r = 0; r < 8; ++r) orow[r] = fmaxf(acc[r], 0.0f);
    }
}

// ---------------------------------------------------------------------------
extern "C" void kernel_launch(void* const* d_in, const int* in_sizes, int n_in,
                              void* d_out, int out_size, void* d_ws, size_t ws_size,
                              hipStream_t stream) {
    const float* inp = (const float*)d_in[0];   // [B, N, 3] fp32
    const float* pnt = (const float*)d_in[1];   // [B, M, 3] fp32

    const int B = 4;
    const int N = in_sizes[0] / (B * 3);        // 8192
    const int M = in_sizes[1] / (B * 3);        // 8192

    float4* qpack = (float4*)d_ws;
    float4* ppack = qpack + (size_t)B * N;

    const int tq = B * N;
    const int tp = B * M;
    nn_pack_queries<<<(tq + 255) / 256, 256, 0, stream>>>(inp, qpack, tq);
    nn_pack_points <<<(tp + 255) / 256, 256, 0, stream>>>(pnt, ppack, tp);

    const int blocksPerBatch = N / 128;         // 128 queries per 256-thread block
    nn_wmma_kernel<<<B * blocksPerBatch, 256, 0, stream>>>(
        qpack, ppack, (float*)d_out, N, M, blocksPerBatch);
}